// SGC_47107201303130
// MI455X (gfx1250) — compile-verified
//
#include <hip/hip_runtime.h>

typedef __attribute__((ext_vector_type(2))) float v2f;
typedef __attribute__((ext_vector_type(8))) float v8f;

#define N_NODES 100000
#define N_EDGES 1600000
#define F_IN    128
#define F_OUT   64
#define M_TILES (N_NODES / 16)   // 6250, exact

// ---------------- degree / normalization ----------------

__global__ void deg_init_kernel(float* __restrict__ deg) {
    int i = blockIdx.x * blockDim.x + threadIdx.x;
    if (i < N_NODES) deg[i] = 1.0f;   // self-loop contributes 1
}

__global__ void deg_count_kernel(const int* __restrict__ dst, float* __restrict__ deg) {
    int e = blockIdx.x * blockDim.x + threadIdx.x;
    if (e < N_EDGES) atomicAdd(&deg[dst[e]], 1.0f);
}

__global__ void dinv_kernel(float* __restrict__ deg) {
    int i = blockIdx.x * blockDim.x + threadIdx.x;
    if (i < N_NODES) deg[i] = __frsqrt_rn(deg[i]);   // deg >= 1 always (self loop)
}

// ---------------- propagation hop ----------------
// Initialize y with the self-loop term: y[i] = dinv[i]^2 * x[i]
__global__ void hop_init_kernel(const float* __restrict__ dinv,
                                const float* __restrict__ xin,
                                float* __restrict__ y) {
    long long tid = (long long)blockIdx.x * blockDim.x + threadIdx.x;
    int i = (int)(tid >> 5);
    if (i >= N_NODES) return;
    int c = (int)(tid & 31);
    float w = dinv[i]; w = w * w;
    const float4 xv = *(const float4*)(xin + (size_t)i * F_IN + c * 4);
    float4 r; r.x = w * xv.x; r.y = w * xv.y; r.z = w * xv.z; r.w = w * xv.w;
    *(float4*)(y + (size_t)i * F_IN + c * 4) = r;
}

// One wave (32 lanes) per edge; lane c handles features [4c, 4c+4).
// Gather x[src] is a coalesced 512B read per edge; scatter via f32 atomics (L2-resident).
__global__ void hop_edge_kernel(const int* __restrict__ src,
                                const int* __restrict__ dst,
                                const float* __restrict__ dinv,
                                const float* __restrict__ xin,
                                float* __restrict__ y) {
    long long tid = (long long)blockIdx.x * blockDim.x + threadIdx.x;
    int e = (int)(tid >> 5);
    if (e >= N_EDGES) return;
    int c = (int)(tid & 31);
    int s = src[e], d = dst[e];
    float w = dinv[s] * dinv[d];
    const float4 xv = *(const float4*)(xin + (size_t)s * F_IN + c * 4);
    float* yp = y + (size_t)d * F_IN + c * 4;
    atomicAdd(yp + 0, w * xv.x);
    atomicAdd(yp + 1, w * xv.y);
    atomicAdd(yp + 2, w * xv.z);
    atomicAdd(yp + 3, w * xv.w);
}

// ---------------- final projection: out = X @ W + b via V_WMMA_F32_16X16X4_F32 ----------------
// One wave per 16-row M-tile; 4 accumulators cover all 64 output columns,
// reusing the A fragment across N-tiles. K loop: 128 / 4 = 32 WMMAs per tile-col.
//
// 32-bit layouts (ISA 7.12.2): A: M=lane%16, K=kbase + v + 2*(lane/16)
//                              B: N=lane%16, K=kbase + v + 2*(lane/16)
//                              C/D: M=v + 8*(lane/16), N=lane%16
__global__ void gemm_wmma_kernel(const float* __restrict__ X,
                                 const float* __restrict__ W,
                                 const float* __restrict__ bias,
                                 float* __restrict__ out) {
    const int wave   = (blockIdx.x * blockDim.x + threadIdx.x) >> 5;
    const int lane   = threadIdx.x & 31;
    if (wave >= M_TILES) return;          // wave-uniform exit: EXEC stays all-1s for live waves

    const int lmod   = lane & 15;
    const int khalf  = (lane >> 4) << 1;  // 0 or 2
    const int row    = wave * 16 + lmod;

    v8f acc[4];
    #pragma unroll
    for (int nt = 0; nt < 4; ++nt) {
        float bv = bias[nt * 16 + lmod];
        #pragma unroll
        for (int v = 0; v < 8; ++v) acc[nt][v] = bv;
    }

    const float* xr = X + (size_t)row * F_IN;
    #pragma unroll 4
    for (int k = 0; k < F_IN; k += 4) {
        v2f a;
        a.x = xr[k + khalf + 0];
        a.y = xr[k + khalf + 1];
        #pragma unroll
        for (int nt = 0; nt < 4; ++nt) {
            v2f b;
            const int ncol = nt * 16 + lmod;
            b.x = W[(size_t)(k + khalf + 0) * F_OUT + ncol];
            b.y = W[(size_t)(k + khalf + 1) * F_OUT + ncol];
            acc[nt] = __builtin_amdgcn_wmma_f32_16x16x4_f32(
                false, a, false, b, (short)0, acc[nt], false, false);
        }
    }

    #pragma unroll
    for (int nt = 0; nt < 4; ++nt) {
        #pragma unroll
        for (int v = 0; v < 8; ++v) {
            int m = wave * 16 + v + 8 * (lane >> 4);
            out[(size_t)m * F_OUT + nt * 16 + lmod] = acc[nt][v];
        }
    }
}

// ---------------- launcher ----------------

extern "C" void kernel_launch(void* const* d_in, const int* in_sizes, int n_in,
                              void* d_out, int out_size, void* d_ws, size_t ws_size,
                              hipStream_t stream) {
    const float* x  = (const float*)d_in[0];           // [100000,128]
    const int*   ei = (const int*)d_in[1];             // [2,1600000]
    const float* W  = (const float*)d_in[2];           // [128,64]
    const float* b  = (const float*)d_in[3];           // [64]
    const int* src = ei;
    const int* dst = ei + N_EDGES;

    char* ws = (char*)d_ws;
    float* deg  = (float*)ws;                                     // 400 KB (becomes dinv)
    float* buf1 = (float*)(ws + (1u << 19));                      // 51.2 MB
    float* buf2 = (float*)(ws + (1u << 19) + (size_t)N_NODES * F_IN * sizeof(float));

    const int B = 256;

    // degree -> dinv (in place)
    deg_init_kernel<<<(N_NODES + B - 1) / B, B, 0, stream>>>(deg);
    deg_count_kernel<<<(N_EDGES + B - 1) / B, B, 0, stream>>>(dst, deg);
    dinv_kernel<<<(N_NODES + B - 1) / B, B, 0, stream>>>(deg);

    const int initGrid = (N_NODES * 32 + B - 1) / B;   // 12500
    const int edgeGrid = (int)(((long long)N_EDGES * 32 + B - 1) / B); // 200000

    // hop 1: x -> buf1
    hop_init_kernel<<<initGrid, B, 0, stream>>>(deg, x, buf1);
    hop_edge_kernel<<<edgeGrid, B, 0, stream>>>(src, dst, deg, x, buf1);

    // hop 2: buf1 -> buf2
    hop_init_kernel<<<initGrid, B, 0, stream>>>(deg, buf1, buf2);
    hop_edge_kernel<<<edgeGrid, B, 0, stream>>>(src, dst, deg, buf1, buf2);

    // projection: buf2 @ W + b -> out  (fp32 WMMA)
    const int wavesNeeded = M_TILES;                   // 6250
    const int wavesPerBlk = B / 32;                    // 8
    gemm_wmma_kernel<<<(wavesNeeded + wavesPerBlk - 1) / wavesPerBlk, B, 0, stream>>>(
        buf2, W, b, (float*)d_out);
}